// DynamicConvolutionAttention_78838419685746
// MI455X (gfx1250) — compile-verified
//
#include <hip/hip_runtime.h>
#include <math.h>

#define B_SZ   256
#define T_SZ   2048
#define RNN    1024
#define HID    128
#define SCH    8
#define SK     21
#define DCH    8
#define DK     21
#define PLEN   11

// workspace layout (in floats)
#define PF_OFF 0                       // 11 floats (prior filter, flipped)
#define M_OFF  64                      // 16 x 128  (concat [U;T] transposed)
#define G_OFF  (64 + 16*HID)           // 256 x 168 dynamic kernels

typedef __attribute__((ext_vector_type(2))) float v2f;
typedef __attribute__((ext_vector_type(8))) float v8f;

// branchless tanh: v_exp_f32 + v_rcp_f32 + fma; no EXEC churn, no div fixup.
// tanh(x) = 1 - 2/(exp(2x)+1); saturates correctly for |x| large.
__device__ __forceinline__ float fast_tanh(float x) {
  float e = __expf(2.0f * x);
  float r = __builtin_amdgcn_rcpf(e + 1.0f);   // raw v_rcp_f32
  return fmaf(-2.0f, r, 1.0f);
}

// ---------------------------------------------------------------- prep ----
__global__ void dca_prep(const float* __restrict__ U_w,
                         const float* __restrict__ T_w,
                         float* __restrict__ ws) {
  int tid = threadIdx.x;
  if (tid < PLEN) {
    // beta-binomial pmf, flipped (Pf[j] = pmf[PLEN-1-j])
    float n = (float)(PLEN - 1);
    float k = (float)(PLEN - 1 - tid);
    float a = 0.1f, b = 0.9f;
    float logp = lgammaf(n + 1.0f) - lgammaf(k + 1.0f) - lgammaf(n - k + 1.0f)
               + lgammaf(k + a) + lgammaf(n - k + b) - lgammaf(n + a + b)
               - (lgammaf(a) + lgammaf(b) - lgammaf(a + b));
    ws[PF_OFF + tid] = expf(logp);
  }
  // M[k][n] = (k<8) ? U_w[n][k] : T_w[n][k-8]   (B-matrix of the fused GEMM)
  for (int idx = tid; idx < 16 * HID; idx += blockDim.x) {
    int k = idx / HID, n = idx % HID;
    float v = (k < SCH) ? U_w[n * SCH + k] : T_w[n * DCH + (k - SCH)];
    ws[M_OFF + k * HID + n] = v;
  }
}

// ------------------------------------------------ dynamic kernels G ------
// h = tanh(s @ W^T + b)  [B,128];  G = h @ V^T  [B,168]
__global__ void dca_dyn(const float* __restrict__ s,
                        const float* __restrict__ W_w,
                        const float* __restrict__ W_b,
                        const float* __restrict__ V_w,
                        float* __restrict__ ws) {
  __shared__ float s_sh[RNN];
  __shared__ float h_sh[HID];
  int b = blockIdx.x, tid = threadIdx.x;   // 128 threads
  for (int i = tid; i < RNN; i += 128) s_sh[i] = s[b * RNN + i];
  __syncthreads();
  float acc = W_b[tid];
  const float* wrow = W_w + (size_t)tid * RNN;
  for (int i = 0; i < RNN; ++i) acc = fmaf(s_sh[i], wrow[i], acc);
  h_sh[tid] = tanhf(acc);                  // precise: feeds dynamic kernels
  __syncthreads();
  float* G = ws + G_OFF + b * (DCH * DK);
  for (int c = tid; c < DCH * DK; c += 128) {
    const float* vrow = V_w + c * HID;
    float g = 0.0f;
    for (int j = 0; j < HID; ++j) g = fmaf(h_sh[j], vrow[j], g);
    G[c] = g;
  }
}

// ---------------------------------------------------------- main fused ---
// One wave handles 16 t-positions: convs -> x[16,16] -> WMMA f32 16x16x4
// chained over K -> tanh -> dot(v) reduction -> + log prior -> e[b,t].
// All small operands staged in LDS; branchless tanh inside the WMMA loop.
#define WPB 8
__global__ void dca_main(const float* __restrict__ alpha,
                         const float* __restrict__ F_w,
                         const float* __restrict__ T_b,
                         const float* __restrict__ v_w,
                         const float* __restrict__ ws,
                         float* __restrict__ e_out) {
  __shared__ float M_sh[16 * HID];      // B-matrix of fused GEMM (16x128)
  __shared__ float W_sh[16 * SK];       // conv rows: k<8 -> F_w, k>=8 -> G[b]
  __shared__ float Tb_sh[HID];
  __shared__ float vw_sh[HID];
  __shared__ float Pf_sh[16];
  __shared__ float red[WPB][32][8];

  int tid  = threadIdx.x;
  int lane = tid & 31;
  int wave = tid >> 5;
  int b    = blockIdx.x >> 4;            // 16 tile-groups per batch row
  int grp  = blockIdx.x & 15;
  int t0   = (grp * WPB + wave) * 16;
  int tloc = lane & 15;
  int tt   = t0 + tloc;
  bool hi  = lane >= 16;

  // ---- cooperative staging (whole block shares batch row b) ----
  for (int i = tid; i < 16 * HID; i += 256) M_sh[i] = ws[M_OFF + i];
  for (int i = tid; i < SCH * SK; i += 256) W_sh[i] = F_w[i];
  for (int i = tid; i < DCH * DK; i += 256)
    W_sh[SCH * SK + i] = ws[G_OFF + b * (DCH * DK) + i];
  if (tid < HID) { Tb_sh[tid] = T_b[tid]; vw_sh[tid] = v_w[tid]; }
  if (tid < PLEN) Pf_sh[tid] = ws[PF_OFF + tid];
  __syncthreads();

  const float* arow = alpha + (size_t)b * T_SZ;

  // 21-tap alpha window for this lane's t (zero-padded at edges)
  float aw[21];
#pragma unroll
  for (int j = 0; j < 21; ++j) {
    int idx = tt - 10 + j;
    aw[j] = (idx >= 0 && idx < T_SZ) ? arow[idx] : 0.0f;
  }

  // 8 x-values this lane owns for its A fragments (tap-major, flat regs):
  // lanes 0-15: K = 4*kt + {0,1};  lanes 16-31: K = 4*kt + {2,3}
  int koff = hi ? 2 : 0;
  float x0 = 0.f, x1 = 0.f, x2 = 0.f, x3 = 0.f;
  float x4 = 0.f, x5 = 0.f, x6 = 0.f, x7 = 0.f;
#pragma unroll
  for (int j = 0; j < 21; ++j) {
    float a = aw[j];
    x0 = fmaf(a, W_sh[(0  + koff) * SK + j], x0);
    x1 = fmaf(a, W_sh[(1  + koff) * SK + j], x1);
    x2 = fmaf(a, W_sh[(4  + koff) * SK + j], x2);
    x3 = fmaf(a, W_sh[(5  + koff) * SK + j], x3);
    x4 = fmaf(a, W_sh[(8  + koff) * SK + j], x4);
    x5 = fmaf(a, W_sh[(9  + koff) * SK + j], x5);
    x6 = fmaf(a, W_sh[(12 + koff) * SK + j], x6);
    x7 = fmaf(a, W_sh[(13 + koff) * SK + j], x7);
  }

  // log prior (lanes 0-15, one per t): p = log(clip(conv(alpha, Pf)))
  float p = 0.0f;
  if (!hi) {
#pragma unroll
    for (int j = 0; j < PLEN; ++j) p = fmaf(aw[j], Pf_sh[j], p);
    p = logf(fmaxf(p, 1e-6f));
  }

  // z = x @ M   via 8 N-tiles x 4 chained V_WMMA_F32_16X16X4_F32
  int col = lane & 15;
  float acc[8];
#pragma unroll
  for (int r = 0; r < 8; ++r) acc[r] = 0.0f;

#pragma unroll 1
  for (int nt = 0; nt < 8; ++nt) {
    int n0 = nt * 16;
    float tb = Tb_sh[n0 + col];
    float vv = vw_sh[n0 + col];
    v8f zc = {0.f, 0.f, 0.f, 0.f, 0.f, 0.f, 0.f, 0.f};

    v2f a0; a0.x = x0; a0.y = x1;
    v2f a1; a1.x = x2; a1.y = x3;
    v2f a2; a2.x = x4; a2.y = x5;
    v2f a3; a3.x = x6; a3.y = x7;

    v2f bv;
    bv.x = M_sh[(0  + koff) * HID + n0 + col];
    bv.y = M_sh[(1  + koff) * HID + n0 + col];
    zc = __builtin_amdgcn_wmma_f32_16x16x4_f32(false, a0, false, bv, (short)0, zc, false, false);
    bv.x = M_sh[(4  + koff) * HID + n0 + col];
    bv.y = M_sh[(5  + koff) * HID + n0 + col];
    zc = __builtin_amdgcn_wmma_f32_16x16x4_f32(false, a1, false, bv, (short)0, zc, false, false);
    bv.x = M_sh[(8  + koff) * HID + n0 + col];
    bv.y = M_sh[(9  + koff) * HID + n0 + col];
    zc = __builtin_amdgcn_wmma_f32_16x16x4_f32(false, a2, false, bv, (short)0, zc, false, false);
    bv.x = M_sh[(12 + koff) * HID + n0 + col];
    bv.y = M_sh[(13 + koff) * HID + n0 + col];
    zc = __builtin_amdgcn_wmma_f32_16x16x4_f32(false, a3, false, bv, (short)0, zc, false, false);

    // D layout: lane<16 reg r -> z[t=r,   n=n0+lane]
    //           lane>=16 reg r -> z[t=8+r, n=n0+lane-16]
#pragma unroll
    for (int r = 0; r < 8; ++r) acc[r] += vv * fast_tanh(zc[r] + tb);
  }

#pragma unroll
  for (int r = 0; r < 8; ++r) red[wave][lane][r] = acc[r];
  __syncthreads();

  if (lane < 16) {  // lane == t; it also owns p for this t
    int base = (lane < 8) ? 0 : 16;
    int r    = lane & 7;
    float esum = 0.0f;
#pragma unroll
    for (int l = 0; l < 16; ++l) esum += red[wave][base + l][r];
    e_out[(size_t)b * T_SZ + t0 + lane] = esum + p;
  }
}

// ------------------------------------------------------- softmax (rows) --
__global__ void dca_softmax(float* __restrict__ e) {
  __shared__ float sred[256];
  int b = blockIdx.x, tid = threadIdx.x;
  float* row = e + (size_t)b * T_SZ;
  float vals[8];
  float m = -INFINITY;
#pragma unroll
  for (int j = 0; j < 8; ++j) { vals[j] = row[tid + j * 256]; m = fmaxf(m, vals[j]); }
  sred[tid] = m;
  __syncthreads();
  for (int s = 128; s > 0; s >>= 1) {
    if (tid < s) sred[tid] = fmaxf(sred[tid], sred[tid + s]);
    __syncthreads();
  }
  m = sred[0];
  __syncthreads();
  float sum = 0.0f;
#pragma unroll
  for (int j = 0; j < 8; ++j) { vals[j] = __expf(vals[j] - m); sum += vals[j]; }
  sred[tid] = sum;
  __syncthreads();
  for (int s = 128; s > 0; s >>= 1) {
    if (tid < s) sred[tid] += sred[tid + s];
    __syncthreads();
  }
  float inv = __builtin_amdgcn_rcpf(sred[0]);
#pragma unroll
  for (int j = 0; j < 8; ++j) row[tid + j * 256] = vals[j] * inv;
}

// ------------------------------------------------------------- launch ----
extern "C" void kernel_launch(void* const* d_in, const int* in_sizes, int n_in,
                              void* d_out, int out_size, void* d_ws, size_t ws_size,
                              hipStream_t stream) {
  const float* s     = (const float*)d_in[0];
  const float* alpha = (const float*)d_in[1];
  const float* W_w   = (const float*)d_in[2];
  const float* W_b   = (const float*)d_in[3];
  const float* V_w   = (const float*)d_in[4];
  const float* F_w   = (const float*)d_in[5];
  const float* U_w   = (const float*)d_in[6];
  const float* T_w   = (const float*)d_in[7];
  const float* T_b   = (const float*)d_in[8];
  const float* v_w   = (const float*)d_in[9];
  float* out = (float*)d_out;
  float* ws  = (float*)d_ws;

  dca_prep<<<1, 256, 0, stream>>>(U_w, T_w, ws);
  dca_dyn<<<B_SZ, 128, 0, stream>>>(s, W_w, W_b, V_w, ws);
  dca_main<<<B_SZ * 16, 256, 0, stream>>>(alpha, F_w, T_b, v_w, ws, out);
  dca_softmax<<<B_SZ, 256, 0, stream>>>(out);
}